// PNAAggregator_38723425140760
// MI455X (gfx1250) — compile-verified
//
#include <hip/hip_runtime.h>
#include <math.h>

typedef float v2f __attribute__((ext_vector_type(2)));
typedef float v8f __attribute__((ext_vector_type(8)));

#define LDS_STRIDE 196  // 192 + 4 pad: lane m hits bank (4m + k) -> conflict-free

// ---------------- CSR build ----------------

__global__ void k_zero_i32(int* p, int n) {
    int i = blockIdx.x * 256 + threadIdx.x;
    if (i < n) p[i] = 0;
}

__global__ void k_count(const int* __restrict__ row, int* __restrict__ deg, int E) {
    int e = blockIdx.x * 256 + threadIdx.x;
    if (e < E) atomicAdd(&deg[row[e]], 1);
}

// per-block exclusive scan of deg -> offsets (partial), block totals -> bsums
__global__ void k_scan_block(const int* __restrict__ deg, int* __restrict__ offsets,
                             int* __restrict__ bsums, int N) {
    __shared__ int s[256];
    int i = blockIdx.x * 256 + threadIdx.x;
    int v = (i < N) ? deg[i] : 0;
    s[threadIdx.x] = v;
    __syncthreads();
    for (int off = 1; off < 256; off <<= 1) {
        int t = (threadIdx.x >= off) ? s[threadIdx.x - off] : 0;
        __syncthreads();
        s[threadIdx.x] += t;
        __syncthreads();
    }
    if (i < N) offsets[i] = s[threadIdx.x] - v;       // exclusive
    if (threadIdx.x == 255) bsums[blockIdx.x] = s[255]; // block total
}

// single-block exclusive scan of the block sums (handles any nb via chunking)
__global__ void k_scan_sums(int* __restrict__ bsums, int nb) {
    __shared__ int s[512];
    int carry = 0;
    for (int base = 0; base < nb; base += 512) {
        int i = base + threadIdx.x;
        int v = (i < nb) ? bsums[i] : 0;
        s[threadIdx.x] = v;
        __syncthreads();
        for (int off = 1; off < 512; off <<= 1) {
            int t = (threadIdx.x >= off) ? s[threadIdx.x - off] : 0;
            __syncthreads();
            s[threadIdx.x] += t;
            __syncthreads();
        }
        if (i < nb) bsums[i] = carry + s[threadIdx.x] - v; // exclusive + carry
        int total = s[511];
        __syncthreads();
        carry += total;
    }
}

__global__ void k_scan_add(int* __restrict__ offsets, const int* __restrict__ bsums,
                           int* __restrict__ cursor, int N, int E) {
    int i = blockIdx.x * 256 + threadIdx.x;
    if (i < N) {
        int off = offsets[i] + bsums[blockIdx.x];
        offsets[i] = off;
        cursor[i]  = off;
    }
    if (blockIdx.x == 0 && threadIdx.x == 0) offsets[N] = E;
}

__global__ void k_fill(const int* __restrict__ row, const int* __restrict__ col,
                       int* __restrict__ cursor, int* __restrict__ csr_src, int E) {
    int e = blockIdx.x * 256 + threadIdx.x;
    if (e < E) {
        int pos = atomicAdd(&cursor[row[e]], 1);
        csr_src[pos] = col[e];
    }
}

// ---------------- fused aggregate (regs->LDS) + WMMA GEMM + tanh ----------------
// block: 128 threads (4 wave32), 16 nodes per block.
// stage 1: 8 threads/node, 8 channels/thread -> sum/max/min in registers -> 16x192 LDS tile
// stage 2: wave w computes D tile [16 nodes x cols 16w..16w+15], K=192 via 48x wmma f32 16x16x4
__global__ void __launch_bounds__(128)
k_fused(const int* __restrict__ offsets, const int* __restrict__ csr_src,
        const float* __restrict__ feat, const float* __restrict__ Wm,
        const float* __restrict__ bias, float* __restrict__ out, int N) {
    __shared__ float aggT[16 * LDS_STRIDE];

    const int tid = threadIdx.x;

    // ---- stage 1: aggregation ----
    {
        int nl = tid >> 3;          // node-in-block 0..15
        int t8 = tid & 7;           // channel group 0..7
        int node = blockIdx.x * 16 + nl;

        float sum[8], mx[8], mn[8];
#pragma unroll
        for (int j = 0; j < 8; ++j) { sum[j] = 0.f; mx[j] = -INFINITY; mn[j] = INFINITY; }

        int s0 = 0, s1 = 0;
        if (node < N) { s0 = offsets[node]; s1 = offsets[node + 1]; }
        for (int e = s0; e < s1; ++e) {
            int c = csr_src[e];
            const float4* fp = (const float4*)(feat + (size_t)c * 64 + t8 * 8);
            float4 f0 = fp[0];
            float4 f1 = fp[1];
            float fv[8] = { f0.x, f0.y, f0.z, f0.w, f1.x, f1.y, f1.z, f1.w };
#pragma unroll
            for (int j = 0; j < 8; ++j) {
                sum[j] += fv[j];
                mx[j] = fmaxf(mx[j], fv[j]);
                mn[j] = fminf(mn[j], fv[j]);
            }
        }
        if (s1 == s0) {  // no incoming edges -> max/min aggregate to 0 (matches reference)
#pragma unroll
            for (int j = 0; j < 8; ++j) { mx[j] = 0.f; mn[j] = 0.f; }
        }
        float* base = &aggT[nl * LDS_STRIDE + t8 * 8];
#pragma unroll
        for (int j = 0; j < 8; ++j) {
            base[j]       = sum[j];
            base[64 + j]  = mx[j];
            base[128 + j] = mn[j];
        }
    }
    __syncthreads();

    // ---- stage 2: WMMA GEMM [16 x 192] x [192 x 64] ----
    {
        int w    = tid >> 5;        // wave 0..3 -> output col tile
        int lane = tid & 31;
        int mlow  = lane & 15;      // M (for A) / N (for B,D)
        int khalf = lane >> 4;      // 0: K pair {0,1}, 1: K pair {2,3}

        int n = w * 16 + mlow;      // output column 0..63

        v8f acc = {};
#pragma unroll 8
        for (int ks = 0; ks < 48; ++ks) {
            int k0 = ks * 4 + khalf * 2;
            v2f a, b;
            a.x = aggT[mlow * LDS_STRIDE + k0];
            a.y = aggT[mlow * LDS_STRIDE + k0 + 1];
            b.x = Wm[(size_t)k0 * 64 + n];
            b.y = Wm[(size_t)(k0 + 1) * 64 + n];
            acc = __builtin_amdgcn_wmma_f32_16x16x4_f32(
                /*neg_a=*/false, a, /*neg_b=*/false, b,
                /*c_mod=*/(short)0, acc, /*reuse_a=*/false, /*reuse_b=*/false);
        }

        float bn = bias[n];
#pragma unroll
        for (int v = 0; v < 8; ++v) {
            int m   = v + khalf * 8;
            int rw  = blockIdx.x * 16 + m;
            if (rw < N) out[(size_t)rw * 64 + n] = tanhf(acc[v] + bn);
        }
    }
}

// ---------------- host side ----------------

extern "C" void kernel_launch(void* const* d_in, const int* in_sizes, int n_in,
                              void* d_out, int out_size, void* d_ws, size_t ws_size,
                              hipStream_t stream) {
    const int*   row  = (const int*)d_in[0];
    const int*   col  = (const int*)d_in[1];
    const float* feat = (const float*)d_in[2];
    const float* Wm   = (const float*)d_in[3];
    const float* bias = (const float*)d_in[4];
    float*       out  = (float*)d_out;

    const int E = in_sizes[0];
    const int N = in_sizes[2] / 64;     // D = 64
    const int nb = (N + 255) / 256;

    // workspace layout (ints): offsets[N+1] | cursor[N] | bsums[nb] | csr_src[E]
    int* offsets = (int*)d_ws;
    int* cursor  = offsets + (N + 1);
    int* bsums   = cursor + N;
    int* csr     = bsums + nb;

    // 1) deg = 0 (use cursor as deg)
    k_zero_i32<<<(N + 255) / 256, 256, 0, stream>>>(cursor, N);
    // 2) count degrees
    k_count<<<(E + 255) / 256, 256, 0, stream>>>(row, cursor, E);
    // 3) exclusive scan -> offsets, cursor copy
    k_scan_block<<<nb, 256, 0, stream>>>(cursor, offsets, bsums, N);
    k_scan_sums<<<1, 512, 0, stream>>>(bsums, nb);
    k_scan_add<<<nb, 256, 0, stream>>>(offsets, bsums, cursor, N, E);
    // 4) fill CSR (store source node per slot)
    k_fill<<<(E + 255) / 256, 256, 0, stream>>>(row, col, cursor, csr, E);
    // 5) fused aggregation + GEMM + bias + tanh
    k_fused<<<(N + 15) / 16, 128, 0, stream>>>(offsets, csr, feat, Wm, bias, out, N);
}